// BasicEuclideanDistModel_6038724018287
// MI455X (gfx1250) — compile-verified
//
#include <hip/hip_runtime.h>
#include <math.h>

// ---------------------------------------------------------------------------
// BasicEuclideanDistModel on MI455X (gfx1250, wave32)
//   out = N_EVENTS*beta - sum_e ||dz + dv*t_e + EPS||
//         - dt * sum_{p,r} exp(beta - ||dz_p + dv_p*t_mid_r + EPS||)
// Memory-bound gather/reduce: ~96 MB of coalesced streams (u,v,t) + L2-resident
// 1.6 MB gather tables. Deterministic 2-level reduction (no float atomics):
// per-wave reduce via V_WMMA_F32_16X16X4_F32 (B = ones), block partials in d_ws,
// single-block finalize. Raw v_sqrt_f32 / v_exp_f32 (inputs are normal-range,
// ~1-2 ulp, far below tolerance over the reductions).
// ---------------------------------------------------------------------------

#define EPSF   1e-6f
#define LOG2E  1.44269504088896340736f
#define RIEMANN 10
#define NB_EV 1024
#define NB_NE 512

typedef __attribute__((ext_vector_type(2))) float v2f;
typedef __attribute__((ext_vector_type(8))) float v8f;

// Wave32 sum via WMMA f32 16x16x4:
//  A (16x4): lane L<16 -> (M=L,K=0..1 in vgpr0/1); lane L>=16 -> (M=L-16,K=2..3).
//  Put x in A-vgpr0, 0 in A-vgpr1; B = all ones => D[m,n] = x_m + x_{m+16}.
//  D layout: vgpr j: lanes 0-15 -> M=j, lanes 16-31 -> M=8+j. Summing a lane's
//  8 D regs gives sum of rows 0-7 (lanes<16) or rows 8-15 (lanes>=16);
//  xor-shuffle by 16 and add -> full wave sum in every lane.
__device__ __forceinline__ float wave_reduce32(float x) {
    v2f a; a[0] = x;    a[1] = 0.0f;
    v2f b; b[0] = 1.0f; b[1] = 1.0f;
    v8f c = {};
    c = __builtin_amdgcn_wmma_f32_16x16x4_f32(
        /*neg_a=*/false, a, /*neg_b=*/false, b,
        /*c_mod=*/(short)0, c, /*reuse_a=*/false, /*reuse_b=*/false);
    float s = c[0] + c[1] + c[2] + c[3] + c[4] + c[5] + c[6] + c[7];
    s += __shfl_xor(s, 16, 32);
    return s;
}

// Deterministic 256-thread block reduce; valid result in thread 0.
__device__ __forceinline__ float block_reduce256(float x) {
    __shared__ float red[8];
    float w = wave_reduce32(x);          // EXEC all-ones here (no divergence yet)
    unsigned lane = threadIdx.x & 31u;
    unsigned wid  = threadIdx.x >> 5;
    if (lane == 0) red[wid] = w;
    __syncthreads();
    float s = 0.0f;
    if (threadIdx.x == 0) {
#pragma unroll
        for (int i = 0; i < 8; ++i) s += red[i];
    }
    __syncthreads();                     // allow safe reuse of `red`
    return s;
}

// ---- event term: sum_e sqrt((dzx+dvx*t+eps)^2 + (dzy+dvy*t+eps)^2) ---------
__global__ __launch_bounds__(256)
void ev_kernel(const float2* __restrict__ z0, const float2* __restrict__ v0,
               const int* __restrict__ u, const int* __restrict__ v,
               const float* __restrict__ et, int n, float* __restrict__ partial) {
    int tid    = blockIdx.x * blockDim.x + threadIdx.x;
    int stride = gridDim.x * blockDim.x;
    float acc = 0.0f;
    for (int i = tid; i < n; i += stride) {
        // stream-ahead prefetch (speculative global_prefetch_b8; OOB is dropped)
        __builtin_prefetch(&u[i + 4 * stride], 0, 0);
        __builtin_prefetch(&v[i + 4 * stride], 0, 0);
        __builtin_prefetch(&et[i + 4 * stride], 0, 0);
        int    a = u[i], bb = v[i];
        float  t = et[i];
        float2 za = z0[a], zb = z0[bb];       // L2-resident gathers (b64)
        float2 va = v0[a], vb = v0[bb];
        float dx = (za.x - zb.x) + (va.x - vb.x) * t + EPSF;
        float dy = (za.y - zb.y) + (va.y - vb.y) * t + EPSF;
        acc += __builtin_amdgcn_sqrtf(dx * dx + dy * dy);   // raw v_sqrt_f32
    }
    float s = block_reduce256(acc);
    if (threadIdx.x == 0) partial[blockIdx.x] = s;
}

// ---- non-event term: sum_{p,r} exp(beta - d(p, t_mid_r)) -------------------
__global__ __launch_bounds__(256)
void ne_kernel(const float2* __restrict__ z0, const float2* __restrict__ v0,
               const int* __restrict__ nu, const int* __restrict__ nv,
               const float* __restrict__ beta, const float* __restrict__ t0p,
               const float* __restrict__ tnp, int n, float* __restrict__ partial) {
    const float betav = beta[0];
    const float t0v   = t0p[0];
    const float dt    = (tnp[0] - t0v) / (float)RIEMANN;
    int tid    = blockIdx.x * blockDim.x + threadIdx.x;
    int stride = gridDim.x * blockDim.x;
    float acc = 0.0f;
    for (int i = tid; i < n; i += stride) {
        int    a = nu[i], bb = nv[i];
        float2 za = z0[a], zb = z0[bb];
        float2 va = v0[a], vb = v0[bb];
        float dzx = za.x - zb.x, dzy = za.y - zb.y;
        float dvx = va.x - vb.x, dvy = va.y - vb.y;
#pragma unroll
        for (int r = 0; r < RIEMANN; ++r) {
            float tm = t0v + ((float)r + 0.5f) * dt;
            float dx = dzx + dvx * tm + EPSF;
            float dy = dzy + dvy * tm + EPSF;
            float d  = __builtin_amdgcn_sqrtf(dx * dx + dy * dy); // v_sqrt_f32
            acc += __builtin_amdgcn_exp2f((betav - d) * LOG2E);   // v_exp_f32
        }
    }
    float s = block_reduce256(acc);
    if (threadIdx.x == 0) partial[blockIdx.x] = s;
}

// ---- finalize: out = n_events*beta - sumEv - dt*sumNE ----------------------
__global__ __launch_bounds__(256)
void fin_kernel(const float* __restrict__ pev, int nev,
                const float* __restrict__ pne, int nne,
                const float* __restrict__ beta, const float* __restrict__ t0p,
                const float* __restrict__ tnp, int n_events,
                float* __restrict__ out) {
    float a = 0.0f, b = 0.0f;
    for (int i = threadIdx.x; i < nev; i += 256) a += pev[i];
    for (int i = threadIdx.x; i < nne; i += 256) b += pne[i];
    float sumEv = block_reduce256(a);
    float sumNe = block_reduce256(b);
    if (threadIdx.x == 0) {
        float dt = (tnp[0] - t0p[0]) / (float)RIEMANN;
        out[0] = (float)n_events * beta[0] - sumEv - dt * sumNe;
    }
}

extern "C" void kernel_launch(void* const* d_in, const int* in_sizes, int n_in,
                              void* d_out, int out_size, void* d_ws, size_t ws_size,
                              hipStream_t stream) {
    const float*  beta = (const float*)d_in[0];
    const float2* z0   = (const float2*)d_in[1];
    const float2* v0   = (const float2*)d_in[2];
    const int*    u    = (const int*)d_in[3];
    const int*    v    = (const int*)d_in[4];
    const float*  et   = (const float*)d_in[5];
    const int*    nu   = (const int*)d_in[6];
    const int*    nv   = (const int*)d_in[7];
    const float*  t0   = (const float*)d_in[8];
    const float*  tn   = (const float*)d_in[9];

    const int n_events = in_sizes[3];
    const int n_pairs  = in_sizes[6];

    float* ws_ev = (float*)d_ws;          // NB_EV partials
    float* ws_ne = ws_ev + NB_EV;         // NB_NE partials (6 KB total)

    ev_kernel<<<NB_EV, 256, 0, stream>>>(z0, v0, u, v, et, n_events, ws_ev);
    ne_kernel<<<NB_NE, 256, 0, stream>>>(z0, v0, nu, nv, beta, t0, tn, n_pairs, ws_ne);
    fin_kernel<<<1, 256, 0, stream>>>(ws_ev, NB_EV, ws_ne, NB_NE,
                                      beta, t0, tn, n_events, (float*)d_out);
}